// ConcatCritic_566935683177
// MI455X (gfx1250) — compile-verified
//
#include <hip/hip_runtime.h>
#include <hip/hip_bf16.h>

typedef __attribute__((ext_vector_type(2))) float v2f;
typedef __attribute__((ext_vector_type(8))) float v8f;

#define LSTR 260   // LDS row stride in floats: 256 + 4 pad -> conflict-free banks

// ---------------------------------------------------------------------------
// Kernel 1: hx[a][g] = sum_d x[a][d]*W0[g][d] + b0[g] ; hy[b][g] = sum_d y[b][d]*W0[g][128+d]
// 2*512*256 threads, dot over 128 with float4.
// ---------------------------------------------------------------------------
__global__ __launch_bounds__(256) void precompute_hxy(
    const float* __restrict__ x, const float* __restrict__ y,
    const float* __restrict__ W0, const float* __restrict__ b0,
    float* __restrict__ hx, float* __restrict__ hy) {
  int t = blockIdx.x * 256 + threadIdx.x;          // 0 .. 262143
  int sel = t >> 17;                               // 0 -> hx, 1 -> hy
  int i = t & 131071;
  int r = i >> 8;                                  // 0..511
  int g = i & 255;                                 // 0..255
  const float4* xr = (const float4*)((sel ? y : x) + r * 128);
  const float4* wr = (const float4*)(W0 + g * 256 + sel * 128);
  float acc = 0.f;
#pragma unroll 8
  for (int k = 0; k < 32; ++k) {
    float4 a = xr[k], w = wr[k];
    acc += a.x * w.x + a.y * w.y + a.z * w.z + a.w * w.w;
  }
  if (sel) hy[i] = acc;
  else     hx[i] = acc + b0[g];                    // fold b0 into hx
}

// ---------------------------------------------------------------------------
// Kernel 2: pack W1/W2 into WMMA f32 B-operand lane order.
// B (4x16, K x N): lane<16 holds (K=k0,k0+1 @ N=lane), lane>=16 holds (K=k0+2,k0+3).
// Two N-tiles packed per float4 so the GEMM loop does b128 loads.
// dst[((kt*8 + p)*32 + lane)] = { W[32p+lo][c], W[32p+lo][c+1], W[32p+16+lo][c], W[32p+16+lo][c+1] }
//   with c = kt*4 + 2*hi.
// ---------------------------------------------------------------------------
__global__ __launch_bounds__(256) void pack_weights(
    const float* __restrict__ W1, const float* __restrict__ W2,
    float* __restrict__ wp1, float* __restrict__ wp2) {
  int t = blockIdx.x * 256 + threadIdx.x;          // 0 .. 32767
  int w = t >> 14;                                 // which weight matrix
  int i = t & 16383;                               // 64 kt * 8 p * 32 lanes
  int lane = i & 31;
  int p  = (i >> 5) & 7;
  int kt = i >> 8;
  int lo = lane & 15, hi = lane >> 4;
  int g0 = 32 * p + lo;
  int c  = kt * 4 + 2 * hi;
  const float* W = w ? W2 : W1;
  float* dst = w ? wp2 : wp1;
  float2 a = *(const float2*)(W + g0 * 256 + c);
  float2 b = *(const float2*)(W + (g0 + 16) * 256 + c);
  float4 o; o.x = a.x; o.y = a.y; o.z = b.x; o.w = b.y;
  ((float4*)dst)[i] = o;
}

// ---------------------------------------------------------------------------
// One 16x256 @ 256x256 GEMM pass: A from this wave's LDS tile, B from packed
// weights, 16 N-tile f32 accumulators in registers (128 VGPRs).
// ---------------------------------------------------------------------------
__device__ __forceinline__ void mlp_layer(const float* __restrict__ myh,
                                          const float4* __restrict__ wp4,
                                          int lane, int arow, v8f acc[16]) {
  for (int kt = 0; kt < 64; ++kt) {
    // A operand: lane lo holds row M=lo, cols kt*4 + 2*hi + {0,1}  (conflict-free)
    v2f A = *(const v2f*)(myh + arow + kt * 4);
    const float4* wk = wp4 + kt * 256 + lane;      // 256 float4 per kt chunk
#pragma unroll
    for (int p = 0; p < 8; ++p) {
      float4 bq = wk[p * 32];
      v2f B0 = {bq.x, bq.y};
      v2f B1 = {bq.z, bq.w};
      acc[2 * p] = __builtin_amdgcn_wmma_f32_16x16x4_f32(
          false, A, false, B0, (short)0, acc[2 * p], false, false);
      acc[2 * p + 1] = __builtin_amdgcn_wmma_f32_16x16x4_f32(
          false, A, false, B1, (short)0, acc[2 * p + 1], false, false);
    }
  }
}

// ---------------------------------------------------------------------------
// Kernel 3: fused pairwise MLP. Block = 8 wave32s; wave wv owns pairs
// (a = by*8+wv, b = bx*16 + 0..15) = 16 rows. All LDS traffic is wave-private
// (no barriers). Two register-accumulated GEMMs + final W3 dot.
// ---------------------------------------------------------------------------
__global__ __launch_bounds__(256) void fused_critic(
    const float* __restrict__ hx, const float* __restrict__ hy,
    const float* __restrict__ wp1, const float* __restrict__ b1,
    const float* __restrict__ wp2, const float* __restrict__ b2,
    const float* __restrict__ w3, const float* __restrict__ b3,
    float* __restrict__ out) {
  extern __shared__ float lds[];                   // 8 waves * 16 rows * LSTR
  const int lane = threadIdx.x & 31;
  const int wv   = threadIdx.x >> 5;               // wave in block, 0..7
  const int lo = lane & 15, hi = lane >> 4;
  const int a   = blockIdx.y * 8 + wv;             // x-row index
  const int bb0 = blockIdx.x * 16;                 // first y-row index
  float* myh = lds + wv * 16 * LSTR;               // this wave's 16x256 tile

  // ---- stage h0 = relu(hx'[a] + hy[b]) into LDS (b0 already folded in hx) --
  float4 hxv[2];
#pragma unroll
  for (int h = 0; h < 2; ++h)
    hxv[h] = *(const float4*)(hx + a * 256 + h * 128 + lane * 4);
  for (int ib = 0; ib < 16; ++ib) {
#pragma unroll
    for (int h = 0; h < 2; ++h) {
      float4 yv = *(const float4*)(hy + (bb0 + ib) * 256 + h * 128 + lane * 4);
      float4 v;
      v.x = fmaxf(hxv[h].x + yv.x, 0.f);
      v.y = fmaxf(hxv[h].y + yv.y, 0.f);
      v.z = fmaxf(hxv[h].z + yv.z, 0.f);
      v.w = fmaxf(hxv[h].w + yv.w, 0.f);
      *(float4*)(myh + ib * LSTR + h * 128 + lane * 4) = v;
    }
  }

  const int arow = lo * LSTR + 2 * hi;
  const v8f vzero = {};
  v8f acc[16];

  // ---- layer 1: h1 = relu(h0 @ W1^T + b1), written back in-place ----------
#pragma unroll
  for (int n = 0; n < 16; ++n) acc[n] = vzero;
  mlp_layer(myh, (const float4*)wp1, lane, arow, acc);
#pragma unroll
  for (int n = 0; n < 16; ++n) {
    float bv = b1[n * 16 + lo];
#pragma unroll
    for (int v = 0; v < 8; ++v)       // C layout: lane holds (M=v+8*hi, N=n*16+lo)
      myh[(v + 8 * hi) * LSTR + n * 16 + lo] = fmaxf(acc[n][v] + bv, 0.f);
  }

  // ---- layer 2: h2 = relu(h1 @ W2^T + b2), kept in registers --------------
#pragma unroll
  for (int n = 0; n < 16; ++n) acc[n] = vzero;
  mlp_layer(myh, (const float4*)wp2, lane, arow, acc);

  // ---- layer 3: out = relu(h2) . w3 + b3 ----------------------------------
  float r[8];
#pragma unroll
  for (int v = 0; v < 8; ++v) r[v] = 0.f;
#pragma unroll
  for (int n = 0; n < 16; ++n) {
    float bv  = b2[n * 16 + lo];
    float w3v = w3[n * 16 + lo];
#pragma unroll
    for (int v = 0; v < 8; ++v)
      r[v] += fmaxf(acc[n][v] + bv, 0.f) * w3v;
  }
  // reduce each r[v] over the 16 lanes of this half (xor masks 1,2,4,8 stay in-half)
#pragma unroll
  for (int v = 0; v < 8; ++v) {
    r[v] += __shfl_xor(r[v], 1, 32);
    r[v] += __shfl_xor(r[v], 2, 32);
    r[v] += __shfl_xor(r[v], 4, 32);
    r[v] += __shfl_xor(r[v], 8, 32);
  }
  if (lo < 8) {
    float val = r[0];
#pragma unroll
    for (int v = 1; v < 8; ++v) val = (lo == v) ? r[v] : val;   // val = r[lo]
    int mrow = lo + 8 * hi;                                      // row M in tile
    out[(size_t)a * 512 + bb0 + mrow] = val + b3[0];
  }
}

// ---------------------------------------------------------------------------
extern "C" void kernel_launch(void* const* d_in, const int* in_sizes, int n_in,
                              void* d_out, int out_size, void* d_ws, size_t ws_size,
                              hipStream_t stream) {
  const float* x  = (const float*)d_in[0];
  const float* y  = (const float*)d_in[1];
  const float* W0 = (const float*)d_in[2];
  const float* b0 = (const float*)d_in[3];
  const float* W1 = (const float*)d_in[4];
  const float* b1 = (const float*)d_in[5];
  const float* W2 = (const float*)d_in[6];
  const float* b2 = (const float*)d_in[7];
  const float* W3 = (const float*)d_in[8];
  const float* b3 = (const float*)d_in[9];
  float* out = (float*)d_out;

  float* ws  = (float*)d_ws;
  float* hx  = ws;                  // 512*256
  float* hy  = ws + 131072;         // 512*256
  float* wp1 = ws + 262144;         // 256*256 packed
  float* wp2 = ws + 327680;         // 256*256 packed  (total 1.5 MB)

  precompute_hxy<<<1024, 256, 0, stream>>>(x, y, W0, b0, hx, hy);
  pack_weights<<<128, 256, 0, stream>>>(W1, W2, wp1, wp2);

  const int lds_bytes = 8 * 16 * LSTR * 4;         // 133,120 B (< 320 KB/WG)
  hipFuncSetAttribute((const void*)fused_critic,
                      hipFuncAttributeMaxDynamicSharedMemorySize, lds_bytes);
  dim3 grid(32, 64);                               // b-tiles x a-tiles
  fused_critic<<<grid, 256, lds_bytes, stream>>>(hx, hy, wp1, b1, wp2, b2,
                                                 W3, b3, out);
}